// Model_69776038691097
// MI455X (gfx1250) — compile-verified
//
#include <hip/hip_runtime.h>

// WMMA 16x16x4 f32: A = 16x4 (2 VGPRs), B = 4x16 (2 VGPRs), C/D = 16x16 f32 (8 VGPRs)
typedef __attribute__((ext_vector_type(2))) float v2f;
typedef __attribute__((ext_vector_type(8))) float v8f;

#define TILE    16
#define WPB     4                   // waves per block (wave32)
#define THREADS (WPB * 32)
#define JCHUNK  1024                // j rows staged in LDS per pass (16 KB)
#define JSPLIT  8                   // j-range split for occupancy

// Positions are pre-scaled by 1/h = 10 so WMMA directly yields q^2 + eps':
//   A = (-2Xi, -2Yi | -2Zi, 1),  B = (Xj, Yj | Zj, |Xj|^2),  C = |Xi|^2 + 1e-8
//   D[M,N] = |Xi - Xj|^2 + 1e-8 = q^2 + eps'   (eps' = 1e-10 / h^2)
// Kernel weight via branchless M4 identity:
//   w/2 = (1-q)+^3 - 4*(0.5-q)+^3   (the factor 2 is folded into the norm)
__global__ __launch_bounds__(THREADS)
void sph_density_wmma(const float* __restrict__ pos,
                      float* __restrict__ outbuf,
                      int n, int jspan, float scale)
{
    // Per j-row, pre-swizzled B operand: [(X,Y), (Z,|X|^2)], X = 10*x.
    __shared__ float4 ldsB[JCHUNK];

    const int lane = threadIdx.x & 31;
    const int wave = threadIdx.x >> 5;
    const int lm   = lane & 15;          // position within 16-lane half
    const int hi   = lane >> 4;          // 0: lanes 0-15, 1: lanes 16-31

    const int itile = blockIdx.x * WPB + wave;
    const int i0    = itile * TILE;

    const float inv_h = 10.0f;          // 1 / H, H = 0.1

    // ---- A tile (i rows): scaled by 1/h, -2 factor folded; K=3 slot = 1.0 ----
    v2f a;
    {
        const float* p = pos + 3 * (i0 + lm);
        const float x = inv_h * p[0], y = inv_h * p[1], z = inv_h * p[2];
        a.x = hi ? -2.0f * z : -2.0f * x;
        a.y = hi ? 1.0f      : -2.0f * y;
    }

    // ---- C operand: |X_i|^2 + eps' for the 8 rows this lane owns (M = v + 8*hi) ----
    v8f si2e;
#pragma unroll
    for (int v = 0; v < 8; ++v) {
        const float* p = pos + 3 * (i0 + v + 8 * hi);
        const float x = inv_h * p[0], y = inv_h * p[1], z = inv_h * p[2];
        si2e[v] = x * x + y * y + z * z + 1e-8f;
    }

    float acc[8];
#pragma unroll
    for (int v = 0; v < 8; ++v) acc[v] = 0.0f;

    const int j_begin = blockIdx.y * jspan;

    for (int jb = j_begin; jb < j_begin + jspan; jb += JCHUNK) {
        // Stage JCHUNK j-rows (scaled, |X|^2 folded) into LDS in B-operand layout.
        __syncthreads();
        for (int r = threadIdx.x; r < JCHUNK; r += THREADS) {
            const float* p = pos + 3 * (jb + r);
            const float x = inv_h * p[0], y = inv_h * p[1], z = inv_h * p[2];
            ldsB[r] = make_float4(x, y, z, x * x + y * y + z * z);
        }
        __syncthreads();

#pragma unroll 4
        for (int t = 0; t < JCHUNK / TILE; ++t) {
            // Single conflict-free b64 LDS read per lane: (X,Y) or (Z,|X|^2).
            const float2* row = (const float2*)&ldsB[t * TILE + lm];
            const float2 bb = row[hi];
            v2f b;
            b.x = bb.x;
            b.y = bb.y;

            // d[v] = q^2 + eps' for M = v + 8*hi, N = lm
            v8f d = __builtin_amdgcn_wmma_f32_16x16x4_f32(
                        false, a, false, b, (short)0, si2e, false, false);

#pragma unroll
            for (int v = 0; v < 8; ++v) {
                // |d| folds into the sqrt source modifier; guards fp cancellation.
                float qq = __builtin_amdgcn_sqrtf(fabsf(d[v]));
                // Branchless cubic spline: w/2 = (1-q)+^3 - 4*(0.5-q)+^3
                float u  = fmaxf(1.0f - qq, 0.0f);
                float w2 = fmaxf(0.5f - qq, 0.0f);
                float u2 = u * u;
                float v2 = w2 * w2;
                acc[v] = fmaf(u2, u, acc[v]);            // + u^3
                acc[v] = fmaf(v2 * w2, -4.0f, acc[v]);   // - 4*v^3
            }
        }
    }

    // Butterfly-reduce over the 16 lanes of each half (sum over N).
#pragma unroll
    for (int v = 0; v < 8; ++v) {
        float s = acc[v];
        s += __shfl_xor(s, 1, 32);
        s += __shfl_xor(s, 2, 32);
        s += __shfl_xor(s, 4, 32);
        s += __shfl_xor(s, 8, 32);
        acc[v] = s * scale;
    }

    // Lane 0 holds rows i0+0..7, lane 16 holds rows i0+8..15.
    if (lm == 0) {
        float* o = outbuf + (size_t)blockIdx.y * n + i0 + 8 * hi;
#pragma unroll
        for (int v = 0; v < 8; ++v) o[v] = acc[v];
    }
}

__global__ void reduce_partials(const float* __restrict__ part,
                                float* __restrict__ out, int n, float scale)
{
    const int i = blockIdx.x * blockDim.x + threadIdx.x;
    if (i < n) {
        float s = 0.0f;
#pragma unroll
        for (int c = 0; c < JSPLIT; ++c) s += part[(size_t)c * n + i];
        out[i] = s * scale;
    }
}

extern "C" void kernel_launch(void* const* d_in, const int* in_sizes, int n_in,
                              void* d_out, int out_size, void* d_ws, size_t ws_size,
                              hipStream_t stream)
{
    const float* pos = (const float*)d_in[0];
    float* out = (float*)d_out;
    const int n = in_sizes[0] / 3;                      // 8192

    // 2 * MASS * 8 / (pi * h^3)  (the 2 from the branchless spline identity)
    const float norm2 = 2.0f * 8.0f / (3.14159265f * 0.1f * 0.1f * 0.1f);

    const int itiles = n / TILE;                        // 512
    dim3 block(THREADS);

    const size_t need = (size_t)JSPLIT * (size_t)n * sizeof(float);
    if (ws_size >= need && (n % (JSPLIT * JCHUNK)) == 0) {
        // Split path: deterministic partials in workspace, then reduce.
        dim3 grid(itiles / WPB, JSPLIT);
        sph_density_wmma<<<grid, block, 0, stream>>>(
            pos, (float*)d_ws, n, n / JSPLIT, 1.0f);
        reduce_partials<<<(n + 255) / 256, 256, 0, stream>>>(
            (const float*)d_ws, out, n, norm2);
    } else {
        // Fallback: each wave walks the whole j-range, writes d_out directly.
        dim3 grid(itiles / WPB, 1);
        sph_density_wmma<<<grid, block, 0, stream>>>(pos, out, n, n, norm2);
    }
}